// arModel_59811714564165
// MI455X (gfx1250) — compile-verified
//
#include <hip/hip_runtime.h>

// AR(4) model over [B=32, K=64, L=8192] as per-batch GEMM M=64, N=8188, K=256
// using V_WMMA_F32_16X16X4_F32 on gfx1250 (wave32).

typedef float v2f __attribute__((ext_vector_type(2)));
typedef float v4f __attribute__((ext_vector_type(4)));
typedef float v8f __attribute__((ext_vector_type(8)));

constexpr int Bn = 32;        // batch
constexpr int Kc = 64;        // channels
constexpr int Ln = 8192;      // sequence length
constexpr int Pn = 4;         // AR order
constexpr int KJ = Kc * Pn;   // reduction length 256
constexpr int NSTEP = KJ / 4; // 64 WMMA k-steps (one per input channel)
constexpr int TT = 256;       // timesteps per workgroup
constexpr int RS = TT + Pn;   // LDS row stride in dwords (260) -> 1040B pitch
constexpr int CH = RS / 4;    // 65 float4 chunks per LDS row

__launch_bounds__(256, 1)
__global__ void ar_wmma_kernel(const float* __restrict__ x,
                               const float* __restrict__ W,
                               const float* __restrict__ bias,
                               float* __restrict__ out) {
  extern __shared__ float xs[];  // xs[k*RS + i] = x[b, k, t0 - 4 + i]
  const int b   = blockIdx.y;
  const int t0  = blockIdx.x * TT;
  const int tid = threadIdx.x;

  // ---- stage x tile (64 x 260 f32) into LDS as b128 chunks ----
  // chunk ci covers i = 4*ci .. 4*ci+3; gt = t0-4+4*ci is 16B aligned, and the
  // only chunk with gt<0 (t0==0, ci==0) is entirely below 0 -> whole-chunk zero.
  const float* xb = x + (size_t)b * Kc * Ln;
  for (int c = tid; c < Kc * CH; c += 256) {
    int k  = c / CH;
    int ci = c - k * CH;
    int i  = ci * 4;
    int gt = t0 - Pn + i;
    v4f v = {0.0f, 0.0f, 0.0f, 0.0f};
    if (gt >= 0)
      v = __builtin_nontemporal_load((const v4f*)(xb + k * Ln + gt));
    *(v4f*)(xs + k * RS + i) = v;
  }
  __syncthreads();

  const int lane  = tid & 31;
  const int wave  = tid >> 5;
  const int laneM = lane & 15;
  const int hi    = lane >> 4;                  // 0: lanes 0-15, 1: lanes 16-31
  const int o0    = (wave & 3) * 16;            // 16-channel output block
  const int thalf = (wave >> 2) * (TT / 2);     // 0 or 128: time half

  // ---- preload A fragments: W[o0..o0+15, :], 16x16x4 f32 A layout ----
  // A vgpr0 = W[o0+laneM, 4s + 2*hi], vgpr1 = next column -> one b64 per step.
  // W is 64KB and read by every workgroup: default (RT) policy keeps it in L2.
  const float* wp = W + (o0 + laneM) * KJ + hi * 2;
  v2f a[NSTEP];
#pragma unroll
  for (int s = 0; s < NSTEP; ++s)
    a[s] = *(const v2f*)(wp + 4 * s);           // 8B-aligned (even dword offs)

  // ---- 16-wide time sub-tiles ----
#pragma unroll 1
  for (int st = 0; st < TT / 2; st += 16) {
    const int tl = thalf + st;                  // local t offset of sub-tile

    // accumulator seeded with bias: C row M=r (lanes 0-15) / M=8+r (16-31)
    v8f acc;
#pragma unroll
    for (int r = 0; r < 8; ++r)
      acc[r] = bias[o0 + r + hi * 8];

    // B fragment: vgpr0 holds rows j={0,2} (half-wave split), vgpr1 = j+1 ->
    // adjacent dword. Per-lane base fixed within the sub-tile; +RS per k-step.
    const float* bp = xs + tl + laneM + 2 * hi;
#pragma unroll
    for (int s = 0; s < NSTEP; ++s) {
      v2f bf;
      bf[0] = bp[s * RS];
      bf[1] = bp[s * RS + 1];
      acc = __builtin_amdgcn_wmma_f32_16x16x4_f32(
          false, a[s], false, bf, (short)0, acc, false, false);
    }

    // ---- store D tile; mask t<4 to exact zero per reference ----
    // lanes 0-15 write 64B of row o0+r, lanes 16-31 write 64B of row o0+8+r.
    const int tglob = t0 + tl + laneM;
#pragma unroll
    for (int r = 0; r < 8; ++r) {
      float v = (tglob < Pn) ? 0.0f : acc[r];
      __builtin_nontemporal_store(
          v, out + ((size_t)b * Kc + (o0 + r + hi * 8)) * Ln + tglob);
    }
  }
}

extern "C" void kernel_launch(void* const* d_in, const int* in_sizes, int n_in,
                              void* d_out, int out_size, void* d_ws, size_t ws_size,
                              hipStream_t stream) {
  const float* x    = (const float*)d_in[0];  // [32, 64, 8192]
  const float* W    = (const float*)d_in[1];  // [64, 256]
  const float* bias = (const float*)d_in[2];  // [64]
  float* out        = (float*)d_out;          // [32, 64, 8192]

  const size_t lds_bytes = (size_t)Kc * RS * sizeof(float);  // 66,560 B
  (void)hipFuncSetAttribute((const void*)ar_wmma_kernel,
                            hipFuncAttributeMaxDynamicSharedMemorySize,
                            (int)lds_bytes);

  dim3 grid(Ln / TT, Bn);   // (32, 32)
  dim3 block(256);          // 8 wave32s
  ar_wmma_kernel<<<grid, block, lds_bytes, stream>>>(x, W, bias, out);
}